// LIFSpike_75299366633687
// MI455X (gfx1250) — compile-verified
//
#include <hip/hip_runtime.h>

// Clang ext-vector so __builtin_nontemporal_{load,store} accept it directly
// (HIP's float4 is a struct and is rejected by those builtins).
typedef __attribute__((ext_vector_type(4))) float v4f;

#define TAU 0.25f

// One thread = one spatial element = one float4 (all 4 timesteps, contiguous
// last axis). 128-bit NT load -> fully unrolled 4-step LIF recurrence in
// registers -> 128-bit NT store. Streaming, single-use data: bypass temporal
// caching (working set 268MB > 192MB L2).
__global__ __launch_bounds__(256) void lif_spike_fwd_kernel(
    const v4f* __restrict__ x,   // [n] packed (elem, 4 steps)
    const float* __restrict__ w, // scalar parameter
    v4f* __restrict__ out,       // [n]
    int n)
{
    int i = blockIdx.x * blockDim.x + threadIdx.x;
    if (i >= n) return;

    // Uniform threshold; per-thread tanh is hidden under memory latency and
    // avoids a serializing prologue kernel inside the captured graph.
    const float thre = tanhf(w[0]);

    v4f xt = __builtin_nontemporal_load(x + i);

    float u, o;
    v4f r;

    // t = 0: u0 = 0, o0 = 0  ->  u = x0
    u = xt.x;
    o = (u - thre) > 0.0f ? 1.0f : 0.0f;
    r.x = o;

    // t = 1..3: u = TAU*u*(1-o) + x_t ; o = step(u - thre)
    u = TAU * u * (1.0f - o) + xt.y;
    o = (u - thre) > 0.0f ? 1.0f : 0.0f;
    r.y = o;

    u = TAU * u * (1.0f - o) + xt.z;
    o = (u - thre) > 0.0f ? 1.0f : 0.0f;
    r.z = o;

    u = TAU * u * (1.0f - o) + xt.w;
    o = (u - thre) > 0.0f ? 1.0f : 0.0f;
    r.w = o;

    __builtin_nontemporal_store(r, out + i);
}

extern "C" void kernel_launch(void* const* d_in, const int* in_sizes, int n_in,
                              void* d_out, int out_size, void* d_ws, size_t ws_size,
                              hipStream_t stream) {
    (void)n_in; (void)d_ws; (void)ws_size; (void)out_size;

    const float* x = (const float*)d_in[0];  // (64,128,32,32,4) f32, flat
    const float* w = (const float*)d_in[1];  // scalar parameter
    float* out = (float*)d_out;

    // Each vector element covers STEPS=4 contiguous floats.
    int n = in_sizes[0] / 4;  // 8,388,608 spatial elements

    const int block = 256;
    const int grid = (n + block - 1) / block;

    lif_spike_fwd_kernel<<<grid, block, 0, stream>>>(
        (const v4f*)x, w, (v4f*)out, n);
}